// dynamic_filter_channel_26018911879615
// MI455X (gfx1250) — compile-verified
//
#include <hip/hip_runtime.h>
#include <math.h>

typedef float v2f __attribute__((ext_vector_type(2)));
typedef float v4f __attribute__((ext_vector_type(4)));
typedef float v8f __attribute__((ext_vector_type(8)));
typedef int   v4i __attribute__((vector_size(16)));   // matches builtin param type

#define N_BATCH 8
#define C_CH    128
#define H_DIM   128
#define W_DIM   128
#define GROUP   8
#define CG      (C_CH / GROUP)   // 16
#define GK2     72               // GROUP * 3 * 3
#define GK2P    80               // padded to 5 WMMA M-tiles
#define HW      (H_DIM * W_DIM)  // 16384
#define IMG_ELEMS (N_BATCH * C_CH * H_DIM * W_DIM)

// ---------------------------------------------------------------------------
// Async global->LDS copy (CDNA5 GLOBAL_LOAD_ASYNC_TO_LDS_B128, ASYNCcnt).
// Builtin signature (from hipcc diagnostic): param0 = v4i* in global AS.
// Falls back to a plain 16B copy if the builtin is unavailable.
// ---------------------------------------------------------------------------
#if defined(__has_builtin)
#if __has_builtin(__builtin_amdgcn_global_load_async_to_lds_b128)
#define HAVE_ASYNC_LDS 1
#endif
#endif

__device__ __forceinline__ void copy16_g2l(const void* g, void* l) {
#ifdef HAVE_ASYNC_LDS
    __builtin_amdgcn_global_load_async_to_lds_b128(
        (__attribute__((address_space(1))) v4i*)g,
        (__attribute__((address_space(3))) v4i*)l, 0, 0);
#else
    *(v4f*)l = *(const v4f*)g;
#endif
}

__device__ __forceinline__ void wait_async_copies() {
#ifdef HAVE_ASYNC_LDS
#if __has_builtin(__builtin_amdgcn_s_wait_asynccnt)
    __builtin_amdgcn_s_wait_asynccnt(0);
#else
    asm volatile("s_wait_asynccnt 0x0" ::: "memory");
#endif
#endif
}

// ---------------------------------------------------------------------------
// Kernel 1: adaptive avg pool (n,c,h,w) -> pooled (n*c)
// 1024 blocks x 256 threads; each block reduces one 128x128 plane (16384 f32).
// Regular-temporal loads on purpose: x (64 MiB) stays resident in the 192 MB
// L2 for the main pass.
// ---------------------------------------------------------------------------
__launch_bounds__(256)
__global__ void pool_kernel(const float* __restrict__ x, float* __restrict__ pooled) {
    const int nc = blockIdx.x;                    // 0..1023
    const v4f* p = (const v4f*)(x + (size_t)nc * HW);
    float s = 0.f;
    for (int i = threadIdx.x; i < HW / 4; i += 256) {
        v4f v = p[i];
        s += v.x + v.y + v.z + v.w;
    }
    for (int off = 16; off > 0; off >>= 1)
        s += __shfl_down(s, off, 32);
    __shared__ float red[8];
    if ((threadIdx.x & 31) == 0) red[threadIdx.x >> 5] = s;
    __syncthreads();
    if (threadIdx.x == 0) {
        float t = 0.f;
        #pragma unroll
        for (int i = 0; i < 8; ++i) t += red[i];
        pooled[nc] = t * (1.0f / (float)HW);
    }
}

// ---------------------------------------------------------------------------
// Kernel 2: filter math — two WMMA f32 GEMMs + sigmoid gate + BN + softmax.
//   GEMM1: lf(72x8) = w_conv(72x128) @ pooled^T(128x8)   K=128 (32 wmma steps)
//   GEMM2: g (72x8) = w_gate(72x72)  @ lf(72x8)          K=72  (18 wmma steps)
// One workgroup, 5 waves; wave w owns M-tile rows [16w,16w+16).
// All operands pre-staged in zero-padded LDS -> predicate-free inner loops:
// 2x ds_load_b64 + v_wmma_f32_16x16x4_f32 per K-step.
// ---------------------------------------------------------------------------
__launch_bounds__(160)
__global__ void filter_kernel(const float* __restrict__ pooled,
                              const float* __restrict__ w_conv,
                              const float* __restrict__ w_gate,
                              const float* __restrict__ bn_gamma,
                              const float* __restrict__ bn_beta,
                              const float* __restrict__ bn_mean,
                              const float* __restrict__ bn_var,
                              float* __restrict__ filt) {
    __shared__ float sConv[GK2P * C_CH];   // 80x128, rows 72..79 = 0 (40960 B)
    __shared__ float sGate[GK2P * GK2];    // 80x72,  rows 72..79 = 0 (23040 B)
    __shared__ float sPool[16 * C_CH];     // 16x128, rows 8..15  = 0 ( 8192 B)
    __shared__ float lfT [16][GK2P];       // lf transposed: [n][o]
    __shared__ float lf2T[16][GK2];        // gated+BN, transposed: [n][o]

    const int tid  = threadIdx.x;
    const int wave = tid >> 5;          // 0..4
    const int lane = tid & 31;
    const int half = lane >> 4;         // 0 | 1
    const int l16  = lane & 15;
    const int m0   = wave * 16;
    const int mA   = m0 + l16;          // A-matrix row this lane feeds

    // ---- stage operands into LDS (async B128 copies), zero the pad rows ----
    {
        v4f* s4;
        // w_conv: 72*128 = 9216 floats = 2304 x 16B
        for (int i = tid; i < (GK2 * C_CH) / 4; i += 160)
            copy16_g2l((const v4f*)w_conv + i, (v4f*)sConv + i);
        // w_gate: 72*72 = 5184 floats = 1296 x 16B (row stride 72 == contiguous)
        for (int i = tid; i < (GK2 * GK2) / 4; i += 160)
            copy16_g2l((const v4f*)w_gate + i, (v4f*)sGate + i);
        // pooled: 8*128 = 1024 floats = 256 x 16B
        for (int i = tid; i < (N_BATCH * C_CH) / 4; i += 160)
            copy16_g2l((const v4f*)pooled + i, (v4f*)sPool + i);
        // zero padding (plain LDS stores)
        const v4f z = {0.f, 0.f, 0.f, 0.f};
        s4 = (v4f*)(sConv + GK2 * C_CH);
        for (int i = tid; i < ((GK2P - GK2) * C_CH) / 4; i += 160) s4[i] = z;
        s4 = (v4f*)(sGate + GK2 * GK2);
        for (int i = tid; i < ((GK2P - GK2) * GK2) / 4; i += 160) s4[i] = z;
        s4 = (v4f*)(sPool + N_BATCH * C_CH);
        for (int i = tid; i < (N_BATCH * C_CH) / 4; i += 160) s4[i] = z;
    }
    wait_async_copies();
    __syncthreads();

    // ---- GEMM1: lf = w_conv @ pooled^T ----
    v8f acc = {0.f, 0.f, 0.f, 0.f, 0.f, 0.f, 0.f, 0.f};
    #pragma unroll
    for (int k = 0; k < C_CH; k += 4) {
        const int ka = k + half * 2;                       // 8B aligned
        v2f a = *(const v2f*)&sConv[mA  * C_CH + ka];
        v2f b = *(const v2f*)&sPool[l16 * C_CH + ka];
        acc = __builtin_amdgcn_wmma_f32_16x16x4_f32(
                  false, a, false, b, (short)0, acc, false, false);
    }
    // D layout: lane l -> col N = l%16; VGPR r -> M = r + 8*(l>=16)
    #pragma unroll
    for (int r = 0; r < 8; ++r)
        lfT[l16][m0 + half * 8 + r] = acc[r];
    __syncthreads();

    // ---- GEMM2: g = w_gate @ lf ----
    v8f acc2 = {0.f, 0.f, 0.f, 0.f, 0.f, 0.f, 0.f, 0.f};
    #pragma unroll
    for (int k = 0; k < GK2; k += 4) {                     // 72 % 4 == 0
        const int ka = k + half * 2;
        v2f a = *(const v2f*)&sGate[mA * GK2 + ka];
        v2f b = *(const v2f*)&lfT[l16][ka];                // contiguous in K
        acc2 = __builtin_amdgcn_wmma_f32_16x16x4_f32(
                   false, a, false, b, (short)0, acc2, false, false);
    }

    // ---- gate + BN (eval form, eps = 1e-5) ----
    #pragma unroll
    for (int r = 0; r < 8; ++r) {
        const int m = m0 + half * 8 + r;
        if (m < GK2) {
            float v  = lfT[l16][m];
            float gg = acc2[r];
            v = v * (1.0f / (1.0f + expf(-gg)));
            v = (v - bn_mean[m]) * rsqrtf(bn_var[m] + 1e-5f)
                  * bn_gamma[m] + bn_beta[m];
            lf2T[l16][m] = v;
        }
    }
    __syncthreads();

    // ---- softmax over 9 taps per (n, group): 64 independent softmaxes ----
    if (tid < N_BATCH * GROUP) {
        const int n_ = tid >> 3;
        const int g_ = tid & 7;
        float e[9];
        float mx = -3.4e38f;
        #pragma unroll
        for (int kk = 0; kk < 9; ++kk) {
            float t = lf2T[n_][g_ * 9 + kk];
            mx = fmaxf(mx, t);
            e[kk] = t;
        }
        float sum = 0.f;
        #pragma unroll
        for (int kk = 0; kk < 9; ++kk) { e[kk] = expf(e[kk] - mx); sum += e[kk]; }
        const float inv = 1.0f / sum;
        #pragma unroll
        for (int kk = 0; kk < 9; ++kk)
            filt[(n_ * GROUP + g_) * 9 + kk] = e[kk] * inv;
    }
}

// ---------------------------------------------------------------------------
// Kernel 3: fused dynamic 3x3 local filter with reflect pad + residual.
// Block = 256 threads = 8 rows x 32 lanes; each lane computes a float4.
// Grid = n * c * (h/8) = 16384 blocks. Filter weights are block-uniform.
// Outputs stored non-temporal (TH=NT): the 128 MiB write stream must not
// evict the L2-resident x (64 MiB fits in the 192 MB L2 after pooling).
// ---------------------------------------------------------------------------
__launch_bounds__(256)
__global__ void local_filter_kernel(const float* __restrict__ x,
                                    const float* __restrict__ filt,
                                    float* __restrict__ out_low,
                                    float* __restrict__ out_high) {
    const int bid = blockIdx.x;
    const int y8  = bid & 15;                 // 0..15 (row chunk)
    const int c_  = (bid >> 4) & (C_CH - 1);  // 0..127
    const int n_  = bid >> 11;                // 0..7

    const int tid = threadIdx.x;
    const int ty  = tid >> 5;                 // 0..7
    const int tx  = tid & 31;                 // 0..31
    const int y   = y8 * 8 + ty;
    const int x0  = tx * 4;

    // block-uniform 9-tap weights for this (n, group)
    const int g = c_ / CG;
    const float* wf = filt + (n_ * GROUP + g) * 9;
    float w[9];
    #pragma unroll
    for (int kk = 0; kk < 9; ++kk) w[kk] = wf[kk];

    const float* img = x + (size_t)(n_ * C_CH + c_) * HW;

    // reflect-padded row indices (pad = 1)
    const int ym = (y == 0)         ? 1         : y - 1;
    const int yp = (y == H_DIM - 1) ? H_DIM - 2 : y + 1;
    const int rows[3] = { ym, y, yp };

    v4f acc = {0.f, 0.f, 0.f, 0.f};
    v4f center;

    #pragma unroll
    for (int i = 0; i < 3; ++i) {
        const float* rp = img + rows[i] * W_DIM;
        const v4f mid = *(const v4f*)(rp + x0);                 // 16B aligned
        // reflect(-1) = 1 -> mid.y ; reflect(128) = 126 -> mid.z
        const float left  = (tx == 0)  ? mid.y : rp[x0 - 1];
        const float right = (tx == 31) ? mid.z : rp[x0 + 4];
        if (i == 1) center = mid;

        const float wl = w[i * 3 + 0];   // tap j=0 -> input col x-1
        const float wc = w[i * 3 + 1];   // tap j=1 -> input col x
        const float wr = w[i * 3 + 2];   // tap j=2 -> input col x+1
        acc.x += wl * left  + wc * mid.x + wr * mid.y;
        acc.y += wl * mid.x + wc * mid.y + wr * mid.z;
        acc.z += wl * mid.y + wc * mid.z + wr * mid.w;
        acc.w += wl * mid.z + wc * mid.w + wr * right;
    }

    const size_t off = (size_t)(n_ * C_CH + c_) * HW + (size_t)y * W_DIM + x0;
    __builtin_nontemporal_store(acc, (v4f*)(out_low + off));
    v4f hi = center - acc;
    __builtin_nontemporal_store(hi, (v4f*)(out_high + off));
}

// ---------------------------------------------------------------------------
extern "C" void kernel_launch(void* const* d_in, const int* in_sizes, int n_in,
                              void* d_out, int out_size, void* d_ws, size_t ws_size,
                              hipStream_t stream) {
    const float* x        = (const float*)d_in[0];
    const float* w_conv   = (const float*)d_in[1];
    const float* w_gate   = (const float*)d_in[2];
    const float* bn_gamma = (const float*)d_in[3];
    const float* bn_beta  = (const float*)d_in[4];
    const float* bn_mean  = (const float*)d_in[5];
    const float* bn_var   = (const float*)d_in[6];

    float* ws     = (float*)d_ws;
    float* pooled = ws;                      // 8*128 = 1024 floats
    float* filt   = ws + N_BATCH * C_CH;     // 8*8*9 = 576 floats

    float* out_low  = (float*)d_out;
    float* out_high = out_low + IMG_ELEMS;

    pool_kernel<<<N_BATCH * C_CH, 256, 0, stream>>>(x, pooled);
    filter_kernel<<<1, 160, 0, stream>>>(pooled, w_conv, w_gate,
                                         bn_gamma, bn_beta, bn_mean, bn_var, filt);
    local_filter_kernel<<<N_BATCH * C_CH * (H_DIM / 8), 256, 0, stream>>>(
        x, filt, out_low, out_high);
}